// Producefactor_55954833933053
// MI455X (gfx1250) — compile-verified
//
#include <hip/hip_runtime.h>
#include <hip/hip_bf16.h>
#include <math.h>

// ---------------- constants ----------------
#define BB     32
#define NA     4
#define NP     16384
#define CC     256
#define KNN    32
#define HEADS  4
#define HD     64
#define ROWS   (BB*NA)      // 128
#define BNEPS  1e-5f

typedef __attribute__((ext_vector_type(2))) float v2f;
typedef __attribute__((ext_vector_type(8))) float v8f;

// =====================================================================
// Kernel 1: KNN (squared distance, top-32 smallest, stable tie-break)
//           + gather sa_x rows + max-pool over K  -> local_feat [128,256]
// One workgroup per (b, a). Distances cached in LDS (64KB of 320KB/WGP).
// =====================================================================
__global__ __launch_bounds__(256) void knn_gather_kernel(
    const float* __restrict__ a_points,   // [B,NA,3]
    const float* __restrict__ sa_x,       // [B,NP,C]
    const float* __restrict__ sa_xyz,     // [B,NP,3]
    float* __restrict__ LF)               // [ROWS, C]
{
    const int wg = blockIdx.x;            // 0..127
    const int b  = wg >> 2;
    const int t  = threadIdx.x;           // 0..255

    __shared__ float sDist[NP];           // 65536 B
    __shared__ int   sIdxK[KNN];
    __shared__ float sRedV[8];
    __shared__ int   sRedI[8];

    const float ax = a_points[wg*3 + 0];
    const float ay = a_points[wg*3 + 1];
    const float az = a_points[wg*3 + 2];
    const float an = ax*ax + ay*ay + az*az;

    const float* xyz = sa_xyz + (size_t)b * NP * 3;
    for (int n = t; n < NP; n += 256) {
        float px = xyz[n*3+0], py = xyz[n*3+1], pz = xyz[n*3+2];
        float d  = -2.0f*(ax*px + ay*py + az*pz) + an + (px*px + py*py + pz*pz);
        sDist[n] = d;
    }
    __syncthreads();

    for (int kk = 0; kk < KNN; ++kk) {
        float bv = 3.4e38f; int bi = NP;
        for (int n = t; n < NP; n += 256) {
            float v = sDist[n];
            if (v < bv || (v == bv && n < bi)) { bv = v; bi = n; }
        }
        // wave32 reduction (value, then lowest index on ties)
        #pragma unroll
        for (int off = 16; off > 0; off >>= 1) {
            float ov = __shfl_down(bv, (unsigned)off, 32);
            int   oi = __shfl_down(bi, (unsigned)off, 32);
            if (ov < bv || (ov == bv && oi < bi)) { bv = ov; bi = oi; }
        }
        if ((t & 31) == 0) { sRedV[t >> 5] = bv; sRedI[t >> 5] = bi; }
        __syncthreads();
        if (t == 0) {
            float fv = sRedV[0]; int fi = sRedI[0];
            #pragma unroll
            for (int w = 1; w < 8; ++w) {
                float ov = sRedV[w]; int oi = sRedI[w];
                if (ov < fv || (ov == fv && oi < fi)) { fv = ov; fi = oi; }
            }
            sIdxK[kk]  = fi;
            sDist[fi]  = 3.4e38f;   // remove from candidate set
        }
        __syncthreads();
    }

    // gather + max over K neighbors; one channel per thread
    const float* X = sa_x + (size_t)b * NP * CC;
    {
        int c = t;   // CC == 256 == blockDim
        float m = -3.4e38f;
        #pragma unroll 4
        for (int kk = 0; kk < KNN; ++kk)
            m = fmaxf(m, X[(size_t)sIdxK[kk] * CC + c]);
        LF[(size_t)wg * CC + c] = m;
    }
}

// =====================================================================
// Kernel 2: pos_lin = (a - mean_a(a)) @ pos_W^T + pos_b   [128,256]
//           gfeat_lin = a @ g_W^T                          [128,256]
// =====================================================================
__global__ __launch_bounds__(256) void prep_kernel(
    const float* __restrict__ a_points,
    const float* __restrict__ pos_W, const float* __restrict__ pos_b,
    const float* __restrict__ g_W,
    float* __restrict__ pos_lin, float* __restrict__ gfeat_lin)
{
    int i = blockIdx.x * blockDim.x + threadIdx.x;   // 0..32767
    if (i >= ROWS * CC) return;
    int row = i >> 8, ch = i & 255;
    int b = row >> 2;
    const float* ap = a_points + b * NA * 3;
    float gx = 0.25f * (ap[0] + ap[3] + ap[6] + ap[9]);
    float gy = 0.25f * (ap[1] + ap[4] + ap[7] + ap[10]);
    float gz = 0.25f * (ap[2] + ap[5] + ap[8] + ap[11]);
    float x = a_points[row*3+0], y = a_points[row*3+1], z = a_points[row*3+2];
    float rx = x - gx, ry = y - gy, rz = z - gz;
    pos_lin[i]   = rx*pos_W[ch*3+0] + ry*pos_W[ch*3+1] + rz*pos_W[ch*3+2] + pos_b[ch];
    gfeat_lin[i] = x*g_W[ch*3+0] + y*g_W[ch*3+1] + z*g_W[ch*3+2];
}

// =====================================================================
// Kernel 3: BatchNorm statistics (training mode, biased var over 128
//           rows per channel)  ->  scale[c], shift[c]
// =====================================================================
__global__ __launch_bounds__(128) void bn_stats_kernel(
    const float* __restrict__ x, const float* __restrict__ gamma,
    const float* __restrict__ beta, int C,
    float* __restrict__ scale, float* __restrict__ shift)
{
    int ch = blockIdx.x;
    int t  = threadIdx.x;                 // row 0..127
    float v = x[(size_t)t * C + ch];
    float s = v, q = v * v;
    __shared__ float sS[4], sQ[4];
    #pragma unroll
    for (int off = 16; off > 0; off >>= 1) {
        s += __shfl_down(s, (unsigned)off, 32);
        q += __shfl_down(q, (unsigned)off, 32);
    }
    if ((t & 31) == 0) { sS[t >> 5] = s; sQ[t >> 5] = q; }
    __syncthreads();
    if (t == 0) {
        float S = sS[0] + sS[1] + sS[2] + sS[3];
        float Q = sQ[0] + sQ[1] + sQ[2] + sQ[3];
        float m   = S * (1.0f / 128.0f);
        float var = fmaxf(Q * (1.0f / 128.0f) - m * m, 0.0f);
        float sc  = gamma[ch] * rsqrtf(var + BNEPS);
        scale[ch] = sc;
        shift[ch] = beta[ch] - m * sc;
    }
}

// =====================================================================
// Kernel 4 (generic): C[M,N] = A[M,K] @ W[N,K]^T (+bias) using
//   V_WMMA_F32_16X16X4_F32  (fp32-exact matrix core path)
// One wave per 16x16 output tile. M multiple of 16; N masked.
// Out-of-range N lanes: clamp row pointer to a valid row, load
// unconditionally, select-to-zero (v_cndmask) -- no EXEC toggling in
// the inner loop (WMMA requires EXEC all-ones anyway).
// A-frag: lane<16 -> K {0,1}, lane>=16 -> K {2,3}; M = lane&15.
// B-frag: same K split; N = lane&15. D: vgpr v -> m = v + 8*(lane>>4).
// =====================================================================
__global__ __launch_bounds__(32) void gemm_nt_wmma_kernel(
    const float* __restrict__ A, const float* __restrict__ W,
    const float* __restrict__ bias, float* __restrict__ Cout,
    int M, int N, int K)
{
    const int n0   = blockIdx.x * 16;
    const int m0   = blockIdx.y * 16;
    const int lane = threadIdx.x;
    const int half = lane >> 4;           // 0 or 1
    const int l16  = lane & 15;

    const int mA     = m0 + l16;
    const int nB     = n0 + l16;
    const bool nValid = (nB < N);
    const int nSafe  = nValid ? nB : (N - 1);   // clamped, always in-bounds

    const float* Arow = A + (size_t)mA * K + (half << 1);
    const float* Wrow = W + (size_t)nSafe * K + (half << 1);

    v8f acc = {};
    for (int k0 = 0; k0 < K; k0 += 4) {
        v2f a;  a.x = Arow[k0];  a.y = Arow[k0 + 1];
        float w0 = Wrow[k0], w1 = Wrow[k0 + 1];   // unconditional loads
        v2f bb;
        bb.x = nValid ? w0 : 0.0f;                // v_cndmask, no branch
        bb.y = nValid ? w1 : 0.0f;
        acc = __builtin_amdgcn_wmma_f32_16x16x4_f32(
            false, a, false, bb, (short)0, acc, false, false);
    }

    if (nValid) {
        float bv = bias ? bias[nB] : 0.0f;
        #pragma unroll
        for (int v = 0; v < 8; ++v) {
            int m = m0 + v + 8 * half;
            Cout[(size_t)m * N + nB] = acc[v] + bv;
        }
    }
}

// =====================================================================
// Kernel 5: tiny multi-head attention over NA=4 anchors.
// One thread per (b, head): 4x4 scores, softmax, av.
// pos is applied on the fly: pos = pos_lin*posScale + posShift.
// =====================================================================
__global__ __launch_bounds__(128) void attn_kernel(
    const float* __restrict__ qkv,        // [ROWS, 3C]
    const float* __restrict__ pos_lin,    // [ROWS, C]
    const float* __restrict__ posScale, const float* __restrict__ posShift,
    float* __restrict__ av)               // [ROWS, C]
{
    int t = threadIdx.x;                  // 0..127
    int b = t >> 2, h = t & 3;
    int rowBase = b * NA;

    float s[4][4];
    #pragma unroll
    for (int m = 0; m < 4; ++m)
        #pragma unroll
        for (int n = 0; n < 4; ++n) s[m][n] = 0.0f;

    for (int d = 0; d < HD; ++d) {
        int c = h * HD + d;
        float ps = posScale[c], pf = posShift[c];
        float qm[4], kn[4];
        #pragma unroll
        for (int m = 0; m < 4; ++m) {
            int row = rowBase + m;
            float p = pos_lin[row*CC + c] * ps + pf;
            qm[m] = qkv[(size_t)row*(3*CC) + c]        + p;
            kn[m] = qkv[(size_t)row*(3*CC) + CC + c]   + p;
        }
        #pragma unroll
        for (int m = 0; m < 4; ++m)
            #pragma unroll
            for (int n = 0; n < 4; ++n) s[m][n] += qm[m] * kn[n];
    }

    float p[4][4];
    const float inv_sqrt_hd = 0.125f;     // 1/sqrt(64)
    #pragma unroll
    for (int m = 0; m < 4; ++m) {
        float mx = -3.4e38f;
        #pragma unroll
        for (int n = 0; n < 4; ++n) mx = fmaxf(mx, s[m][n] * inv_sqrt_hd);
        float sum = 0.0f;
        #pragma unroll
        for (int n = 0; n < 4; ++n) {
            p[m][n] = __expf(s[m][n] * inv_sqrt_hd - mx);
            sum += p[m][n];
        }
        float inv = 1.0f / sum;
        #pragma unroll
        for (int n = 0; n < 4; ++n) p[m][n] *= inv;
    }

    for (int d = 0; d < HD; ++d) {
        int c = h * HD + d;
        float ps = posScale[c], pf = posShift[c];
        float vn[4];
        #pragma unroll
        for (int n = 0; n < 4; ++n) {
            int row = rowBase + n;
            vn[n] = qkv[(size_t)row*(3*CC) + 2*CC + c] + pos_lin[row*CC + c] * ps + pf;
        }
        #pragma unroll
        for (int m = 0; m < 4; ++m) {
            float o = p[m][0]*vn[0] + p[m][1]*vn[1] + p[m][2]*vn[2] + p[m][3]*vn[3];
            av[(size_t)(rowBase + m)*CC + c] = o;
        }
    }
}

// =====================================================================
// Kernel 6: feat[row, 0:C]   = LF + BN(res_lin)
//           feat[row, C:2C]  = max_a BN(gfeat_lin)[b,a,:]   (broadcast)
// =====================================================================
__global__ __launch_bounds__(256) void feat_kernel(
    const float* __restrict__ LF, const float* __restrict__ res_lin,
    const float* __restrict__ resScale, const float* __restrict__ resShift,
    const float* __restrict__ gfeat_lin,
    const float* __restrict__ gScale, const float* __restrict__ gShift,
    float* __restrict__ feat)             // [ROWS, 2C]
{
    int i = blockIdx.x * blockDim.x + threadIdx.x;
    if (i >= ROWS * CC) return;
    int row = i >> 8, ch = i & 255;
    float r = res_lin[i] * resScale[ch] + resShift[ch];
    feat[(size_t)row * (2*CC) + ch] = LF[i] + r;

    int b = row >> 2;
    float sc = gScale[ch], sh = gShift[ch];
    float m = -3.4e38f;
    #pragma unroll
    for (int a = 0; a < NA; ++a)
        m = fmaxf(m, gfeat_lin[(size_t)(b*NA + a)*CC + ch] * sc + sh);
    feat[(size_t)row * (2*CC) + CC + ch] = m;
}

// =====================================================================
// Kernel 7: out = prob_lin * pScale + pShift   [ROWS, 9]
// =====================================================================
__global__ __launch_bounds__(256) void out_kernel(
    const float* __restrict__ prob_lin,
    const float* __restrict__ pScale, const float* __restrict__ pShift,
    float* __restrict__ out)
{
    int i = blockIdx.x * blockDim.x + threadIdx.x;
    if (i >= ROWS * 9) return;
    int ch = i % 9;
    out[i] = prob_lin[i] * pScale[ch] + pShift[ch];
}

// =====================================================================
// launcher
// =====================================================================
extern "C" void kernel_launch(void* const* d_in, const int* in_sizes, int n_in,
                              void* d_out, int out_size, void* d_ws, size_t ws_size,
                              hipStream_t stream) {
    const float* a_points  = (const float*)d_in[0];
    const float* sa_x      = (const float*)d_in[1];
    const float* sa_xyz    = (const float*)d_in[2];
    /* d_in[3] = xyz_raw : unused by reference */
    const float* g_W       = (const float*)d_in[4];
    const float* g_gamma   = (const float*)d_in[5];
    const float* g_beta    = (const float*)d_in[6];
    const float* qkv_W     = (const float*)d_in[7];
    const float* pos_W     = (const float*)d_in[8];
    const float* pos_b     = (const float*)d_in[9];
    const float* pos_gamma = (const float*)d_in[10];
    const float* pos_beta  = (const float*)d_in[11];
    const float* res_W     = (const float*)d_in[12];
    const float* res_b     = (const float*)d_in[13];
    const float* res_gamma = (const float*)d_in[14];
    const float* res_beta  = (const float*)d_in[15];
    const float* prob_W    = (const float*)d_in[16];
    const float* prob_gamma= (const float*)d_in[17];
    const float* prob_beta = (const float*)d_in[18];
    float* out = (float*)d_out;

    // workspace carve-up (floats)
    float* ws = (float*)d_ws;
    float* LF        = ws;                 // 32768
    float* pos_lin   = LF        + 32768;  // 32768
    float* gfeat_lin = pos_lin   + 32768;  // 32768
    float* qkv       = gfeat_lin + 32768;  // 98304
    float* av        = qkv       + 98304;  // 32768
    float* res_lin   = av        + 32768;  // 32768
    float* feat      = res_lin   + 32768;  // 65536
    float* prob_lin  = feat      + 65536;  // 1152 (pad 2048)
    float* posScale  = prob_lin  + 2048;   // 256
    float* posShift  = posScale  + 256;
    float* gScale    = posShift  + 256;
    float* gShift    = gScale    + 256;
    float* resScale  = gShift    + 256;
    float* resShift  = resScale  + 256;
    float* pScale    = resShift  + 256;    // 16
    float* pShift    = pScale    + 16;     // 16

    // 1) KNN + gather + maxpool -> local_feat
    knn_gather_kernel<<<ROWS, 256, 0, stream>>>(a_points, sa_x, sa_xyz, LF);

    // 2) pos_lin / gfeat_lin
    prep_kernel<<<(ROWS*CC + 255)/256, 256, 0, stream>>>(
        a_points, pos_W, pos_b, g_W, pos_lin, gfeat_lin);

    // 3) BN stats for pos and gfeat
    bn_stats_kernel<<<CC, 128, 0, stream>>>(pos_lin, pos_gamma, pos_beta, CC, posScale, posShift);
    bn_stats_kernel<<<CC, 128, 0, stream>>>(gfeat_lin, g_gamma, g_beta, CC, gScale, gShift);

    // 4) qkv = LF @ qkv_W^T  (M=128, N=768, K=256) via f32 WMMA
    gemm_nt_wmma_kernel<<<dim3(768/16, ROWS/16), 32, 0, stream>>>(
        LF, qkv_W, nullptr, qkv, ROWS, 3*CC, CC);

    // 5) attention -> av
    attn_kernel<<<1, 128, 0, stream>>>(qkv, pos_lin, posScale, posShift, av);

    // 6) res_lin = av @ res_W^T + res_b  (M=128, N=256, K=256)
    gemm_nt_wmma_kernel<<<dim3(CC/16, ROWS/16), 32, 0, stream>>>(
        av, res_W, res_b, res_lin, ROWS, CC, CC);

    // 7) BN stats for res
    bn_stats_kernel<<<CC, 128, 0, stream>>>(res_lin, res_gamma, res_beta, CC, resScale, resShift);

    // 8) feat = [LF + BN(res) , broadcast(max_a BN(gfeat))]
    feat_kernel<<<(ROWS*CC + 255)/256, 256, 0, stream>>>(
        LF, res_lin, resScale, resShift, gfeat_lin, gScale, gShift, feat);

    // 9) prob_lin = feat @ prob_W^T  (M=128, N=9 masked, K=512)
    gemm_nt_wmma_kernel<<<dim3(1, ROWS/16), 32, 0, stream>>>(
        feat, prob_W, nullptr, prob_lin, ROWS, 9, 2*CC);

    // 10) BN stats for prob, 11) final output
    bn_stats_kernel<<<9, 128, 0, stream>>>(prob_lin, prob_gamma, prob_beta, 9, pScale, pShift);
    out_kernel<<<(ROWS*9 + 255)/256, 256, 0, stream>>>(prob_lin, pScale, pShift, out);
}